// KernelRnn_44513041056215
// MI455X (gfx1250) — compile-verified
//
#include <hip/hip_runtime.h>
#include <math.h>
#include <stdint.h>

// ---------------------------------------------------------------------------
// KernelRnn slow_update, fully collapsed to one streaming pass.
//   h = sum_c a_c*chem_c + sum_c w_c*tanh(sum_d K_cd*chem_d)
//     + sum_r qm_r*mean_r + sum_r qvi_r*varup_r - sum_r qv_r*mean_r^2
// Memory-bound: 285 MB of HBM traffic -> ~12.3 us floor @ 23.3 TB/s.
// Data path: double-buffered async global->LDS (CDNA5 ASYNCcnt), each thread
// reads back its own LDS slots (no barrier), hardware v_tanh_f32 for the mix.
// ---------------------------------------------------------------------------

#define CC 5
#define RR 14
#define ROWS (CC + 2 * RR)   // 33 input planes
#define BLK 256              // threads per block (8 waves of 32)
#define TPB_TILES 8          // tiles processed per block

// coefficient buffer layout in d_ws (floats):
// [0..13]=qm  [14..27]=qvi(=qv/t)  [28..41]=qv  [42..46]=a  [47..51]=w  [52..76]=K

__device__ __forceinline__ float dev_tanh(float x) {
#if __has_builtin(__builtin_amdgcn_tanhf)
  return __builtin_amdgcn_tanhf(x);
#elif __has_builtin(__builtin_amdgcn_tanh_f32)
  return __builtin_amdgcn_tanh_f32(x);
#else
  // CDNA5 hardware transcendental; v_nop covers the TRANS-op result hazard
  float r;
  asm("v_tanh_f32 %0, %1\n\tv_nop" : "=v"(r) : "v"(x));
  return r;
#endif
}

// async copy 16B/lane from global to LDS; tracked on ASYNCcnt
__device__ __forceinline__ void async_b128(uint32_t ldsaddr, uint64_t gaddr) {
  asm volatile("global_load_async_to_lds_b128 %0, %1, off"
               :: "v"(ldsaddr), "v"(gaddr)
               : "memory");
}
__device__ __forceinline__ void wait_async_le_rows() {
  asm volatile("s_wait_asynccnt 33" ::: "memory");   // prev stage done, next in flight
}
__device__ __forceinline__ void wait_async_zero() {
  asm volatile("s_wait_asynccnt 0" ::: "memory");
}

// ---------------- setup: fold Q,K_slow,v,y,z,1/t into 77 coefficients -------
__global__ void krnn_setup(const float* __restrict__ Q, const float* __restrict__ Ks,
                           const float* __restrict__ v, const float* __restrict__ y,
                           const float* __restrict__ z, const int* __restrict__ tix,
                           float* __restrict__ coef) {
  if (threadIdx.x != 0 || blockIdx.x != 0) return;
  const float inv_t = 1.0f / (float)tix[0];
  float w[CC], a[CC];
  for (int c = 0; c < CC; ++c) { w[c] = v[c] * z[c]; a[c] = v[c] * y[c]; }
  for (int r = 0; r < RR; ++r) {
    float qm = 0.f, qv = 0.f;
    for (int c = 0; c < CC; ++c) {
      qm += w[c] * Q[c * (2 * RR) + r];
      qv += w[c] * Q[c * (2 * RR) + RR + r];
    }
    coef[r] = qm;
    coef[RR + r] = qv * inv_t;
    coef[2 * RR + r] = qv;
  }
  for (int c = 0; c < CC; ++c) { coef[3 * RR + c] = a[c]; coef[3 * RR + CC + c] = w[c]; }
  for (int i = 0; i < CC * CC; ++i) coef[3 * RR + 2 * CC + i] = Ks[i];
}

// ---------------- main streaming kernel ------------------------------------
__global__ void __launch_bounds__(BLK)
krnn_main(const float* __restrict__ chem, const float* __restrict__ mean,
          const float* __restrict__ varu, const float* __restrict__ coef,
          float* __restrict__ out, int mn, int tilesTotal) {
  extern __shared__ float4 smem[];   // 2 stages * 33 rows * 256 threads * 16B = 264 KB
  const int tid = threadIdx.x;
  const uint32_t ldsBase = (uint32_t)(uintptr_t)smem;
  const uint64_t chB = (uint64_t)(uintptr_t)chem;
  const uint64_t muB = (uint64_t)(uintptr_t)mean;
  const uint64_t vaB = (uint64_t)(uintptr_t)varu;
  const uint64_t rowBytes = (uint64_t)mn * 4ull;

  int t0 = blockIdx.x * TPB_TILES;
  int tEnd = t0 + TPB_TILES;
  if (tEnd > tilesTotal) tEnd = tilesTotal;
  if (t0 >= tEnd) return;

  auto prefetch = [&](int tile, int stage) {
    const uint64_t off = ((uint64_t)tile * BLK + (uint64_t)tid) * 16ull;
    const uint32_t l = ldsBase + (uint32_t)((stage * ROWS * BLK + tid) * 16);
#pragma unroll
    for (int c = 0; c < CC; ++c)
      async_b128(l + (uint32_t)(c * BLK * 16), chB + (uint64_t)c * rowBytes + off);
#pragma unroll
    for (int r = 0; r < RR; ++r)
      async_b128(l + (uint32_t)((CC + r) * BLK * 16), muB + (uint64_t)r * rowBytes + off);
#pragma unroll
    for (int r = 0; r < RR; ++r)
      async_b128(l + (uint32_t)((CC + RR + r) * BLK * 16), vaB + (uint64_t)r * rowBytes + off);
  };

  // start HBM streaming immediately, before the (independent) SMEM coef loads
  prefetch(t0, 0);

  // uniform coefficient loads (constant offsets off a kernarg pointer -> SGPRs)
  float qm[RR], qvi[RR], qv[RR], a5[CC], w5[CC], K5[CC * CC];
#pragma unroll
  for (int r = 0; r < RR; ++r) {
    qm[r] = coef[r]; qvi[r] = coef[RR + r]; qv[r] = coef[2 * RR + r];
  }
#pragma unroll
  for (int c = 0; c < CC; ++c) {
    a5[c] = coef[3 * RR + c]; w5[c] = coef[3 * RR + CC + c];
  }
#pragma unroll
  for (int i = 0; i < CC * CC; ++i) K5[i] = coef[3 * RR + 2 * CC + i];

  for (int t = t0; t < tEnd; ++t) {
    const int s = (t - t0) & 1;
    if (t + 1 < tEnd) { prefetch(t + 1, s ^ 1); wait_async_le_rows(); }
    else              { wait_async_zero(); }

    const int base = s * ROWS * BLK + tid;
    float ax = 0.f, ay = 0.f, az = 0.f, aw = 0.f;

    // update-rule part: qm*mean + qvi*varup - qv*mean^2
#pragma unroll
    for (int r = 0; r < RR; ++r) {
      float4 mu = smem[base + (CC + r) * BLK];
      float4 vv = smem[base + (CC + RR + r) * BLK];
#define RPART(c_)                                                     \
      a##c_ = fmaf(qm[r], mu.c_, a##c_);                              \
      a##c_ = fmaf(qvi[r], vv.c_, a##c_);                             \
      a##c_ = fmaf(-qv[r] * mu.c_, mu.c_, a##c_);
      RPART(x) RPART(y) RPART(z) RPART(w)
#undef RPART
    }

    // chemical part: a_c*chem_c + w_c*tanh(K @ chem)
    float4 ch[CC];
#pragma unroll
    for (int c = 0; c < CC; ++c) ch[c] = smem[base + c * BLK];
#pragma unroll
    for (int c = 0; c < CC; ++c) {
#define APART(c_) a##c_ = fmaf(a5[c], ch[c].c_, a##c_);
      APART(x) APART(y) APART(z) APART(w)
#undef APART
    }
#pragma unroll
    for (int c = 0; c < CC; ++c) {
      float gx = 0.f, gy = 0.f, gz = 0.f, gw = 0.f;
#pragma unroll
      for (int d = 0; d < CC; ++d) {
#define GACC(c_) g##c_ = fmaf(K5[c * CC + d], ch[d].c_, g##c_);
        GACC(x) GACC(y) GACC(z) GACC(w)
#undef GACC
      }
#define TPART(c_) a##c_ = fmaf(w5[c], dev_tanh(g##c_), a##c_);
      TPART(x) TPART(y) TPART(z) TPART(w)
#undef TPART
    }

    const uint64_t q = (uint64_t)t * BLK + (uint64_t)tid;  // float4 index
    float4 o; o.x = ax; o.y = ay; o.z = az; o.w = aw;
    *(float4*)((char*)out + q * 16ull) = o;
  }
}

// ---------------- host entry ------------------------------------------------
extern "C" void kernel_launch(void* const* d_in, const int* in_sizes, int n_in,
                              void* d_out, int out_size, void* d_ws, size_t ws_size,
                              hipStream_t stream) {
  const float* chem = (const float*)d_in[0];
  const float* mean = (const float*)d_in[1];
  const float* varu = (const float*)d_in[2];
  const float* Q    = (const float*)d_in[3];
  const float* Ks   = (const float*)d_in[4];
  const float* v    = (const float*)d_in[5];
  const float* y    = (const float*)d_in[6];
  const float* z    = (const float*)d_in[7];
  const int*   tix  = (const int*)d_in[8];
  float* coef = (float*)d_ws;

  krnn_setup<<<1, 32, 0, stream>>>(Q, Ks, v, y, z, tix, coef);

  const int mn = in_sizes[0] / CC;                 // M*N = 2^21
  const int tilesTotal = mn / (BLK * 4);           // 1024 elements per tile
  const int blocks = (tilesTotal + TPB_TILES - 1) / TPB_TILES;
  const size_t shmem = (size_t)2 * ROWS * BLK * sizeof(float4);  // 264 KB of 320 KB LDS

  krnn_main<<<blocks, BLK, shmem, stream>>>(chem, mean, varu, coef,
                                            (float*)d_out, mn, tilesTotal);
}